// GINModel_75634374083203
// MI455X (gfx1250) — compile-verified
//
#include <hip/hip_runtime.h>

// CDNA5 / gfx1250, wave32. fp32 WMMA 16x16x4 for all GEMMs.
typedef __attribute__((ext_vector_type(2))) float v2f;
typedef __attribute__((ext_vector_type(8))) float v8f;

__global__ __launch_bounds__(64) void zero_out_kernel(float* __restrict__ out, int n) {
  int i = threadIdx.x;
  if (i < n) out[i] = 0.0f;
}

__global__ __launch_bounds__(256) void copy_f4_kernel(const float4* __restrict__ in,
                                                      float4* __restrict__ out, int n4) {
  int i = blockIdx.x * 256 + threadIdx.x;
  if (i < n4) out[i] = in[i];
}

// Edge-parallel segment-sum: agg[dst] += feat[src], D = 4 << LOG2DQ features.
// One thread handles 4 consecutive floats (float4 gather, 4 f32 atomics).
template <int LOG2DQ>
__global__ __launch_bounds__(256) void scatter_add_kernel(const float* __restrict__ feat,
                                                          float* __restrict__ agg,
                                                          const int* __restrict__ ei,
                                                          int nEdges) {
  constexpr int DQ = 1 << LOG2DQ;
  constexpr int D  = DQ * 4;
  int tid   = blockIdx.x * 256 + threadIdx.x;
  int total = nEdges << LOG2DQ;
  if (tid >= total) return;
  int e  = tid >> LOG2DQ;
  int f4 = tid & (DQ - 1);
  int src = ei[e];
  int dst = ei[nEdges + e];
  const float4 v = *(const float4*)(feat + (size_t)src * D + (size_t)f4 * 4);
  float* p = agg + (size_t)dst * D + (size_t)f4 * 4;
  atomicAdd(p + 0, v.x);
  atomicAdd(p + 1, v.y);
  atomicAdd(p + 2, v.z);
  atomicAdd(p + 3, v.w);
}

// C[M,64] = act(A[M,K] @ W[K,64] + bias), optionally column-reduced into out[64].
// Block: 256 threads = 8 waves; each wave owns a 16-row M tile and all 64 cols
// (4 accumulators of 16 cols each). K stepped by 4 via V_WMMA_F32_16X16X4_F32.
//
// Fragment layouts (CDNA5 ISA 7.12.2, wave32):
//   A 16x4 f32 : lanes 0-15 -> M=lane,   {v0,v1}={K0,K1}; lanes 16-31 -> {K2,K3}
//   B 4x16 f32 : lanes 0-15 -> N=lane,   {v0,v1}={K0,K1}; lanes 16-31 -> {K2,K3}
//   C/D 16x16  : lane<16: N=lane,  VGPR r -> M=r ; lane>=16: N=lane-16, M=r+8
//
// Weights are staged into LDS pre-swizzled so that each B fragment
// {W[k][n], W[k+1][n]} is a contiguous 8-byte pair -> single ds_load_b64:
//   wswz[(k>>1)*128 + n*2 + (k&1)] = W[k][n]
template <int K, bool RELU, bool REDUCE>
__global__ __launch_bounds__(256) void gemm_wmma_kernel(const float* __restrict__ A,
                                                        const float* __restrict__ W,
                                                        const float* __restrict__ bias,
                                                        float* __restrict__ out, int M) {
  __shared__ float wswz[K * 64];  // swizzled: (K/2) rows of 128 floats
  __shared__ float blds[64];
  __shared__ float csum[64];

  for (int i = threadIdx.x; i < K * 64; i += 256) {
    int k = i >> 6;
    int n = i & 63;
    wswz[((k >> 1) << 7) + (n << 1) + (k & 1)] = W[i];
  }
  if (threadIdx.x < 64) {
    blds[threadIdx.x] = bias[threadIdx.x];
    if (REDUCE) csum[threadIdx.x] = 0.0f;
  }
  __syncthreads();

  const int wave = threadIdx.x >> 5;
  const int lane = threadIdx.x & 31;
  const int grp  = lane >> 4;   // 0: K0/K1 half, 1: K2/K3 half
  const int idx  = lane & 15;   // row for A, col for B/C/D
  const int tileRow = (blockIdx.x * 8 + wave) * 16;

  if (tileRow < M) {  // M % 16 == 0 -> uniform per-wave, EXEC stays all-ones
    const float* arow = A + (size_t)(tileRow + idx) * K;
    const v2f* wp = (const v2f*)wswz;  // v2f element n at kpair p: wp[p*64 + n]
    v8f acc0 = {}, acc1 = {}, acc2 = {}, acc3 = {};

#pragma unroll
    for (int kb = 0; kb < K; kb += 4) {
      const int ka = kb + 2 * grp;             // even
      const v2f a = *(const v2f*)(arow + ka);  // global_load_b64
      const int base = ((kb >> 1) + grp) * 64 + idx;
      const v2f b0 = wp[base +  0];  // ds_load_b64 (pairable into 2addr_b64)
      const v2f b1 = wp[base + 16];
      const v2f b2 = wp[base + 32];
      const v2f b3 = wp[base + 48];
      acc0 = __builtin_amdgcn_wmma_f32_16x16x4_f32(false, a, false, b0, (short)0, acc0, false, false);
      acc1 = __builtin_amdgcn_wmma_f32_16x16x4_f32(false, a, false, b1, (short)0, acc1, false, false);
      acc2 = __builtin_amdgcn_wmma_f32_16x16x4_f32(false, a, false, b2, (short)0, acc2, false, false);
      acc3 = __builtin_amdgcn_wmma_f32_16x16x4_f32(false, a, false, b3, (short)0, acc3, false, false);
    }

    v8f accv[4] = {acc0, acc1, acc2, acc3};
#pragma unroll
    for (int nt = 0; nt < 4; ++nt) {
      const int col = nt * 16 + idx;
      const float bv = blds[col];
      if (REDUCE) {
        float s = 0.0f;
#pragma unroll
        for (int r = 0; r < 8; ++r) {
          float t = accv[nt][r] + bv;
          if (RELU) t = fmaxf(t, 0.0f);
          s += t;
        }
        atomicAdd(&csum[col], s);  // ds_add_f32
      } else {
#pragma unroll
        for (int r = 0; r < 8; ++r) {
          float t = accv[nt][r] + bv;
          if (RELU) t = fmaxf(t, 0.0f);
          out[(size_t)(tileRow + r + 8 * grp) * 64 + col] = t;
        }
      }
    }
  }

  if (REDUCE) {
    __syncthreads();
    if (threadIdx.x < 64) atomicAdd(&out[threadIdx.x], csum[threadIdx.x]);
  }
}

extern "C" void kernel_launch(void* const* d_in, const int* in_sizes, int n_in,
                              void* d_out, int out_size, void* d_ws, size_t ws_size,
                              hipStream_t stream) {
  const float* x   = (const float*)d_in[0];
  const int*   ei  = (const int*)d_in[1];
  const float* w1a = (const float*)d_in[2];
  const float* b1a = (const float*)d_in[3];
  const float* w1b = (const float*)d_in[4];
  const float* b1b = (const float*)d_in[5];
  const float* w2a = (const float*)d_in[6];
  const float* b2a = (const float*)d_in[7];
  const float* w2b = (const float*)d_in[8];
  const float* b2b = (const float*)d_in[9];
  float* out = (float*)d_out;

  const int DIN = 128, DH = 64;
  const int M = in_sizes[0] / DIN;  // 100000 nodes
  const int E = in_sizes[1] / 2;    // 1600000 edges

  // Workspace layout (floats): [ agg1: M*128 | h1: M*64 | h2: M*64 ]
  float* agg1 = (float*)d_ws;
  float* h1   = agg1 + (size_t)M * DIN;
  float* h2   = h1 + (size_t)M * DH;
  float* agg2 = agg1;  // agg1 dead after h1 is computed
  float* h3   = h1;    // h1 dead after h2 is computed

  const int mtiles  = (M + 15) / 16;
  const int gblocks = (mtiles + 7) / 8;

  // out = 0 (harness poisons d_out)
  zero_out_kernel<<<1, 64, 0, stream>>>(out, out_size);

  // ---- conv1 aggregation: agg1 = x + segment_sum(x[src], dst) ----
  {
    int n4 = M * DIN / 4;
    copy_f4_kernel<<<(n4 + 255) / 256, 256, 0, stream>>>((const float4*)x, (float4*)agg1, n4);
  }
  {
    int total = E * (DIN / 4);
    scatter_add_kernel<5><<<(total + 255) / 256, 256, 0, stream>>>(x, agg1, ei, E);
  }

  // h1 = relu(agg1 @ w1a + b1a)   [K=128]
  gemm_wmma_kernel<128, true, false><<<gblocks, 256, 0, stream>>>(agg1, w1a, b1a, h1, M);
  // h2 = relu(h1 @ w1b + b1b)     [K=64]
  gemm_wmma_kernel<64, true, false><<<gblocks, 256, 0, stream>>>(h1, w1b, b1b, h2, M);

  // ---- conv2 aggregation: agg2 = h2 + segment_sum(h2[src], dst) ----
  {
    int n4 = M * DH / 4;
    copy_f4_kernel<<<(n4 + 255) / 256, 256, 0, stream>>>((const float4*)h2, (float4*)agg2, n4);
  }
  {
    int total = E * (DH / 4);
    scatter_add_kernel<4><<<(total + 255) / 256, 256, 0, stream>>>(h2, agg2, ei, E);
  }

  // h3 = relu(agg2 @ w2a + b2a)
  gemm_wmma_kernel<64, true, false><<<gblocks, 256, 0, stream>>>(agg2, w2a, b2a, h3, M);
  // out += colsum(h3 @ w2b + b2b)  (fused global add-pool)
  gemm_wmma_kernel<64, false, true><<<gblocks, 256, 0, stream>>>(h3, w2b, b2b, out, M);
}